// IDWT_22771916603490
// MI455X (gfx1250) — compile-verified
//
#include <hip/hip_runtime.h>

// IDWT (inverse 2D Haar) for B=16, C=64, h=w=128  ->  out [16,64,256,256] fp32.
// Banded Haar matrices collapse the three matmuls into a per-element 2x2
// butterfly; kernel is pure streaming (512 MiB traffic, ~22us floor @23.3TB/s).
// Data path: gfx1250 async global->LDS b128 copies (ASYNCcnt) with software
// double buffering, ds_load_b128 readback, non-temporal b128 stores.

typedef float v4f __attribute__((ext_vector_type(4)));
typedef int   v4i __attribute__((vector_size(16)));   // matches builtin param type

#define AS1 __attribute__((address_space(1)))
#define AS3 __attribute__((address_space(3)))

#if __has_builtin(__builtin_amdgcn_s_wait_asynccnt)
#define WAIT_ASYNC(n) __builtin_amdgcn_s_wait_asynccnt(n)
#else
#define WAIT_ASYNC(n) asm volatile("s_wait_asynccnt %0" :: "i"(n) : "memory")
#endif

__device__ __forceinline__ void async_cp16(const float* gsrc, v4f* ldst) {
#if __has_builtin(__builtin_amdgcn_global_load_async_to_lds_b128)
  __builtin_amdgcn_global_load_async_to_lds_b128(
      (AS1 v4i*)(unsigned long long)gsrc,   // global source (16B per lane)
      (AS3 v4i*)ldst,                       // LDS destination (per-lane)
      0, 0);
#else
  unsigned loff = (unsigned)(unsigned long long)(AS3 void*)ldst;
  unsigned long long ga = (unsigned long long)gsrc;
  asm volatile("global_load_async_to_lds_b128 %0, %1, off"
               :: "v"(loff), "v"(ga) : "memory");
#endif
}

__global__ __launch_bounds__(256) void idwt_haar_async(
    const float* __restrict__ pLL, const float* __restrict__ pLH,
    const float* __restrict__ pHL, const float* __restrict__ pHH,
    float* __restrict__ out)
{
  constexpr int TPB  = 256;
  constexpr int ITER = 8;
  // [buffer parity][array][lane] : 2*4*256*16B = 32 KB LDS
  __shared__ v4f smem[2][4][TPB];

  const int tid = threadIdx.x;
  // group index: one float4 of each input per thread-iteration
  const long long g0 = (long long)blockIdx.x * (ITER * TPB) + tid;

  // Prologue: async-stage iteration 0 into buffer 0.
  {
    const long long gb = 4 * g0;
    async_cp16(pLL + gb, &smem[0][0][tid]);
    async_cp16(pLH + gb, &smem[0][1][tid]);
    async_cp16(pHL + gb, &smem[0][2][tid]);
    async_cp16(pHH + gb, &smem[0][3][tid]);
  }

  const float s = 0.7071067811865476f;  // 1/sqrt(2), matches reference matrices

#pragma unroll
  for (int it = 0; it < ITER; ++it) {
    // Double buffering: prefetch next chunk, then wait only for current one.
    if (it + 1 < ITER) {
      const long long gp = (g0 + (long long)(it + 1) * TPB) * 4;
      const int nb = (it + 1) & 1;
      async_cp16(pLL + gp, &smem[nb][0][tid]);
      async_cp16(pLH + gp, &smem[nb][1][tid]);
      async_cp16(pHL + gp, &smem[nb][2][tid]);
      async_cp16(pHH + gp, &smem[nb][3][tid]);
      WAIT_ASYNC(4);   // the 4 just-issued may remain in flight
    } else {
      WAIT_ASYNC(0);
    }
    asm volatile("" ::: "memory");  // compiler barrier: LDS was written async

    // Each lane reads back only its own LDS slots -> no workgroup barrier.
    const int b = it & 1;
    v4f ll = smem[b][0][tid];
    v4f lh = smem[b][1][tid];
    v4f hl = smem[b][2][tid];
    v4f hh = smem[b][3][tid];

    // Column butterfly (matrix_low_1 / matrix_high_1)
    v4f l0 = s * (ll - lh);   // L at even cols
    v4f l1 = s * (ll + lh);   // L at odd cols
    v4f h0 = s * (hl - hh);   // H at even cols
    v4f h1 = s * (hl + hh);   // H at odd cols
    // Row butterfly (matrix_low_0 / matrix_high_0)
    v4f r0e = s * (l0 - h0);  // row 2i,   even cols
    v4f r0o = s * (l1 - h1);  // row 2i,   odd cols
    v4f r1e = s * (l0 + h0);  // row 2i+1, even cols
    v4f r1o = s * (l1 + h1);  // row 2i+1, odd cols

    // Interleave even/odd columns into contiguous 16B stores.
    v4f r0a = { r0e.x, r0o.x, r0e.y, r0o.y };
    v4f r0b = { r0e.z, r0o.z, r0e.w, r0o.w };
    v4f r1a = { r1e.x, r1o.x, r1e.y, r1o.y };
    v4f r1b = { r1e.z, r1o.z, r1e.w, r1o.w };

    // Decompose group index: 32 float4-quads per row, 128 rows per plane.
    const long long g  = g0 + (long long)it * TPB;
    const int  jq = (int)(g & 31);          // quad within row
    const int  i  = (int)((g >> 5) & 127);  // input row
    const long long p = g >> 12;            // plane (b*C + c), 4096 groups/plane

    float* row0 = out + p * 65536LL + (long long)(2 * i) * 256 + jq * 8;
    float* row1 = row0 + 256;

    // Write-once output: non-temporal, keep the 192MB L2 clean.
    __builtin_nontemporal_store(r0a, (v4f*)row0);
    __builtin_nontemporal_store(r0b, (v4f*)row0 + 1);
    __builtin_nontemporal_store(r1a, (v4f*)row1);
    __builtin_nontemporal_store(r1b, (v4f*)row1 + 1);
  }
}

extern "C" void kernel_launch(void* const* d_in, const int* in_sizes, int n_in,
                              void* d_out, int out_size, void* d_ws, size_t ws_size,
                              hipStream_t stream) {
  const float* LL = (const float*)d_in[0];
  const float* LH = (const float*)d_in[1];
  const float* HL = (const float*)d_in[2];
  const float* HH = (const float*)d_in[3];
  // d_in[4..7] are the banded Haar matrices; their structure is baked into the
  // butterfly (s = 1/sqrt(2)), so they are not read.
  float* out = (float*)d_out;

  const long long n = in_sizes[0];                    // 16*64*128*128 = 16,777,216
  const int blocks  = (int)(n / (4LL * 256 * 8));     // = 2048 (exact division)
  idwt_haar_async<<<blocks, 256, 0, stream>>>(LL, LH, HL, HH, out);
}